// ComirecSA_29377576304916
// MI455X (gfx1250) — compile-verified
//
#include <hip/hip_runtime.h>
#include <hip/hip_bf16.h>

typedef __attribute__((ext_vector_type(2))) float v2f;
typedef __attribute__((ext_vector_type(8))) float v8f;

#define B_   4096
#define S_   50
#define E_   64
#define K_   4
#define V_   100000
#define D_   256
#define NTILES_  (V_ / 16)    // 6250
#define VSPLIT_  8            // V-slices (gridDim.y of scores kernel)
#define VCHUNK_  782          // ceil(6250/8)
#define RPB_     128          // batch rows per block (8 waves x 16 rows)

// ---------------------------------------------------------------------------
// Kernel 1: per-batch attention. One 256-thread block per batch element.
// Computes multi_interest (written to d_out) and "best" vector (to d_ws).
// ---------------------------------------------------------------------------
__global__ __launch_bounds__(256) void attention_kernel(
    const int*   __restrict__ item_seq,   // (B,S)
    const float* __restrict__ mask,       // (B,S)
    const int*   __restrict__ item,       // (B,1)
    const float* __restrict__ emb,        // (V,E)
    const float* __restrict__ W1,         // (E,D)
    const float* __restrict__ W2,         // (D,K)
    float*       __restrict__ out_mi,     // (B,K,E) at d_out
    float*       __restrict__ best)       // (B,E) scratch
{
    __shared__ float s_seq[S_ * E_];      // 12.8 KB
    __shared__ float s_W2[K_ * D_];       // 4 KB, transposed [k][d]
    __shared__ float s_A[S_ * K_];
    __shared__ float s_mi[K_ * E_];
    __shared__ float s_item[E_];
    __shared__ float s_cos[K_];
    __shared__ int   s_idx[S_];
    __shared__ int   s_kidx;

    const int tid  = threadIdx.x;
    const int b    = blockIdx.x;
    const int lane = tid & 31;

    if (tid < S_) s_idx[tid] = item_seq[b * S_ + tid];
    for (int f = tid; f < K_ * D_; f += 256) {
        int d = f & (D_ - 1), k = f >> 8;
        s_W2[f] = W2[d * K_ + k];
    }
    if (tid < S_ * K_) {
        int s = tid >> 2;
        s_A[tid] = -1.0e9f * (1.0f - mask[b * S_ + s]);
    }
    __syncthreads();

    // gather seq_emb into LDS (row 0 of table is zero)
    for (int f = tid; f < S_ * E_; f += 256) {
        int s = f >> 6, e = f & 63;
        int idx = s_idx[s];
        s_seq[f] = idx ? emb[(size_t)idx * E_ + e] : 0.0f;
    }
    __syncthreads();

    // A[s][k] = sum_d tanh(sum_e seq[s][e]*W1[e][d]) * W2[d][k]  (+ mask bias in s_A)
    for (int s = 0; s < S_; ++s) {
        float acc = 0.0f;
        const float* se  = s_seq + s * E_;
        const float* w1c = W1 + tid;               // W1[e][d=tid], coalesced
        #pragma unroll
        for (int e = 0; e < E_; ++e) acc += se[e] * w1c[e * D_];
        float h = tanhf(acc);
        #pragma unroll
        for (int k = 0; k < K_; ++k) {
            float p = h * s_W2[k * D_ + tid];
            for (int off = 16; off >= 1; off >>= 1) p += __shfl_xor(p, off, 32);
            if (lane == 0) atomicAdd(&s_A[s * K_ + k], p);
        }
    }
    __syncthreads();

    // softmax over s for each of the 4 heads
    if (tid < K_) {
        float mx = -3.0e38f;
        for (int s = 0; s < S_; ++s) mx = fmaxf(mx, s_A[s * K_ + tid]);
        float sum = 0.0f;
        for (int s = 0; s < S_; ++s) sum += __expf(s_A[s * K_ + tid] - mx);
        float inv = 1.0f / sum;
        for (int s = 0; s < S_; ++s) s_A[s * K_ + tid] = __expf(s_A[s * K_ + tid] - mx) * inv;
    }
    __syncthreads();

    // multi_interest[k][e] = sum_s A[s][k] * seq[s][e]
    {
        int k = tid >> 6, e = tid & 63;
        float mi = 0.0f;
        for (int s = 0; s < S_; ++s) mi += s_A[s * K_ + k] * s_seq[s * E_ + e];
        out_mi[(size_t)b * (K_ * E_) + tid] = mi;
        s_mi[tid] = mi;
    }
    if (tid < E_) {
        int idx = item[b];
        s_item[tid] = idx ? emb[(size_t)idx * E_ + tid] : 0.0f;
    }
    __syncthreads();

    if (tid < K_) {
        float c = 0.0f;
        for (int e = 0; e < E_; ++e) c += s_mi[tid * E_ + e] * s_item[e];
        s_cos[tid] = c;
    }
    __syncthreads();
    if (tid == 0) {
        int kb = 0; float bv = s_cos[0];
        for (int k = 1; k < K_; ++k) if (s_cos[k] > bv) { bv = s_cos[k]; kb = k; }
        s_kidx = kb;
    }
    __syncthreads();
    if (tid < E_) best[(size_t)b * E_ + tid] = s_mi[s_kidx * E_ + tid];
}

// ---------------------------------------------------------------------------
// Kernel 2: scores = best @ table.T fused with online logsumexp (partials).
// 256-thread block = 8 waves; each wave owns its OWN 16-row M-tile (block
// covers 128 batch rows) and all 8 waves stream the SAME V-slice of table
// tiles, so one L2 fill per tile is broadcast to 8 waves through the WGP$.
// gridDim.y = 8 splits V for parallelism. B-fragments are staged through an
// explicit 16-entry register array off a single incremented base pointer so
// the compiler can clause all 16 loads and stagger the waits against the
// chained WMMAs (round-1 pipeline shape). No LDS, no barriers, no spills.
// ---------------------------------------------------------------------------
__global__ __launch_bounds__(256) void scores_lse_kernel(
    const float* __restrict__ best,       // (B,E)
    const float* __restrict__ emb,        // (V,E)
    float*       __restrict__ part_m,     // (VSPLIT, B)
    float*       __restrict__ part_s)     // (VSPLIT, B)
{
    const int tid   = threadIdx.x;
    const int wave  = tid >> 5;
    const int lane  = tid & 31;
    const int n     = lane & 15;   // WMMA N col / A row slot
    const int khalf = lane >> 4;   // which K-pair this lane supplies
    const int r0    = blockIdx.x * RPB_ + wave * 16;  // this wave's 16 rows
    const int vy    = blockIdx.y;
    const int vstart = vy * VCHUNK_;
    const int vend   = (vstart + VCHUNK_ < NTILES_) ? (vstart + VCHUNK_) : NTILES_;

    // A fragments: best rows r0..r0+15. A layout 16x4 per chunk c:
    // lane<16 -> (K=4c+0,4c+1), lane>=16 -> (K=4c+2,4c+3), row = lane%16
    v2f afrag[16];
    {
        const float* ap = best + (size_t)(r0 + n) * E_ + 2 * khalf;
        #pragma unroll
        for (int c = 0; c < 16; ++c) afrag[c] = *(const v2f*)(ap + 4 * c);
    }

    float mrun[8], srun[8];
    #pragma unroll
    for (int i = 0; i < 8; ++i) { mrun[i] = -3.0e38f; srun[i] = 0.0f; }

    // single incremented base pointer -> constant-offset loads inside the tile
    const float* bp = emb + ((size_t)vstart * 16 + n) * E_ + 2 * khalf;
    const size_t bstride = (size_t)16 * E_;   // floats per vocab tile

    for (int vt = vstart; vt < vend; ++vt, bp += bstride) {
        // prefetch two tiles ahead (global_prefetch_b8; no counter cost)
        if (vt + 2 < vend) __builtin_prefetch(bp + 2 * bstride, 0, 1);

        // stage all 16 B-fragments first: 16 independent clause-able loads
        v2f bf[16];
        #pragma unroll
        for (int c = 0; c < 16; ++c) bf[c] = *(const v2f*)(bp + 4 * c);

        v8f acc = {0.f, 0.f, 0.f, 0.f, 0.f, 0.f, 0.f, 0.f};
        #pragma unroll
        for (int c = 0; c < 16; ++c) {
            acc = __builtin_amdgcn_wmma_f32_16x16x4_f32(
                false, afrag[c], false, bf[c], (short)0, acc, false, false);
        }

        // lane holds scores for column (vt*16+n), rows M = i + 8*khalf
        #pragma unroll
        for (int i = 0; i < 8; ++i) {
            float x = acc[i];
            if (x > mrun[i]) {
                srun[i] = srun[i] * __expf(mrun[i] - x) + 1.0f;
                mrun[i] = x;
            } else {
                srun[i] += __expf(x - mrun[i]);
            }
        }
    }

    // combine across the 16 lanes of each half (they hold disjoint vocab cols)
    #pragma unroll
    for (int i = 0; i < 8; ++i) {
        float m = mrun[i], s = srun[i];
        for (int off = 8; off >= 1; off >>= 1) {
            float om = __shfl_xor(m, off, 32);
            float os = __shfl_xor(s, off, 32);
            float M  = fmaxf(m, om);
            s = s * __expf(m - M) + os * __expf(om - M);
            m = M;
        }
        if (n == 0) {
            int r = r0 + i + 8 * khalf;          // this wave's global row
            part_m[(size_t)vy * B_ + r] = m;
            part_s[(size_t)vy * B_ + r] = s;
        }
    }
}

// ---------------------------------------------------------------------------
// Kernel 2b: merge the 8 V-slice partials per row, apply row-0 zeroing and
// pos_score corrections, emit per-row loss terms (deterministic).
// ---------------------------------------------------------------------------
__global__ __launch_bounds__(256) void lse_finalize_kernel(
    const float* __restrict__ best,       // (B,E)
    const float* __restrict__ emb,        // (V,E)
    const int*   __restrict__ item,       // (B,1)
    const float* __restrict__ part_m,     // (VSPLIT, B)
    const float* __restrict__ part_s,     // (VSPLIT, B)
    float*       __restrict__ loss_terms) // (B,)
{
    const int r = blockIdx.x * 256 + threadIdx.x;
    if (r >= B_) return;

    float M = -3.0e38f, S = 0.0f;
    for (int v = 0; v < VSPLIT_; ++v) {
        float m = part_m[(size_t)v * B_ + r];
        float s = part_s[(size_t)v * B_ + r];
        float Mn = fmaxf(M, m);
        S = S * __expf(M - Mn) + s * __expf(m - Mn);
        M = Mn;
    }
    const float* br = best + (size_t)r * E_;
    int pos = item[r];
    const float* ep = emb + (size_t)pos * E_;
    float s0 = 0.0f, ps = 0.0f;
    for (int e = 0; e < E_; ++e) {
        float bv = br[e];
        s0 += bv * emb[e];   // bogus score vs un-zeroed table row 0
        ps += bv * ep[e];
    }
    S += __expf(-M) - __expf(s0 - M);   // replace row-0 term with exp(0)
    if (pos == 0) ps = 0.0f;
    float log_z = M + __logf(S);
    loss_terms[r] = log_z - ps;
}

// ---------------------------------------------------------------------------
// Kernel 3: deterministic mean of 4096 loss terms -> d_out[B*K*E]
// ---------------------------------------------------------------------------
__global__ __launch_bounds__(256) void loss_reduce_kernel(
    const float* __restrict__ terms, float* __restrict__ out_loss)
{
    __shared__ float red[256];
    int tid = threadIdx.x;
    float s = 0.0f;
    for (int i = tid; i < B_; i += 256) s += terms[i];
    red[tid] = s;
    __syncthreads();
    for (int off = 128; off >= 1; off >>= 1) {
        if (tid < off) red[tid] += red[tid + off];
        __syncthreads();
    }
    if (tid == 0) out_loss[0] = red[0] * (1.0f / (float)B_);
}

extern "C" void kernel_launch(void* const* d_in, const int* in_sizes, int n_in,
                              void* d_out, int out_size, void* d_ws, size_t ws_size,
                              hipStream_t stream) {
    const int*   item_seq = (const int*)  d_in[0];
    const float* mask     = (const float*)d_in[1];
    const int*   item     = (const int*)  d_in[2];
    const float* emb      = (const float*)d_in[3];
    const float* W1       = (const float*)d_in[4];
    const float* W2       = (const float*)d_in[5];
    float* out    = (float*)d_out;
    float* best   = (float*)d_ws;                       // B*E floats (1 MB)
    float* terms  = best  + (size_t)B_ * E_;            // B floats
    float* part_m = terms + B_;                         // VSPLIT*B floats
    float* part_s = part_m + (size_t)VSPLIT_ * B_;      // VSPLIT*B floats

    attention_kernel<<<B_, 256, 0, stream>>>(item_seq, mask, item, emb, W1, W2, out, best);

    dim3 grid2(B_ / RPB_, VSPLIT_);
    scores_lse_kernel<<<grid2, 256, 0, stream>>>(best, emb, part_m, part_s);

    lse_finalize_kernel<<<B_ / 256, 256, 0, stream>>>(best, emb, item, part_m, part_s, terms);
    loss_reduce_kernel<<<1, 256, 0, stream>>>(terms, out + (size_t)B_ * K_ * E_);
}